// CRFSequenceHead_5849745457649
// MI455X (gfx1250) — compile-verified
//
#include <hip/hip_runtime.h>
#include <hip/hip_bf16.h>
#include <stdint.h>

// Problem constants (from reference): B, T, H, L
#define CB 64
#define CT 2048
#define CH 256
#define CL 16
#define CS 32              // timesteps per scan chunk
#define CCH (CT / CS)      // 64 chunks; chunk 0 handled as a vector scan
#define FNEG -1.0e30f      // finite "-inf" (no NaN in exp(a-b))

typedef __attribute__((ext_vector_type(2))) float v2f;
typedef __attribute__((ext_vector_type(8))) float v8f;

// ---------------------------------------------------------------------------
// Stable 16-way logsumexp split across lane halves: each lane supplies 8
// values; partner lane (lane^16) supplies the other 8. Returns lse of all 16.
// ---------------------------------------------------------------------------
__device__ __forceinline__ float lse8_halfmerge(float v0, float v1, float v2, float v3,
                                                float v4, float v5, float v6, float v7) {
    const float m = fmaxf(fmaxf(fmaxf(v0, v1), fmaxf(v2, v3)),
                          fmaxf(fmaxf(v4, v5), fmaxf(v6, v7)));
    const float s = __expf(v0 - m) + __expf(v1 - m) + __expf(v2 - m) + __expf(v3 - m)
                  + __expf(v4 - m) + __expf(v5 - m) + __expf(v6 - m) + __expf(v7 - m);
    const float om = __shfl_xor(m, 16);
    const float os = __shfl_xor(s, 16);
    const float M  = fmaxf(m, om);
    const float S  = s * __expf(m - M) + os * __expf(om - M);
    return M + __logf(S);
}

// ---------------------------------------------------------------------------
// Kernel 1: emissions[b,t,l] = sum_h x[b,t,h] * W[l,h] + bias[l]
// One wave32 per 16-row tile; fp32 V_WMMA_F32_16X16X4_F32, K chained 64x.
// ---------------------------------------------------------------------------
__global__ __launch_bounds__(256)
void crf_emissions_wmma(const float* __restrict__ x, const float* __restrict__ W,
                        const float* __restrict__ bias, float* __restrict__ em) {
    const int lane = threadIdx.x & 31;
    const int wave = threadIdx.x >> 5;
    const int tile = blockIdx.x * 8 + wave;
    const int m0   = tile * 16;
    const int n    = lane & 15;
    const int koff = (lane >> 4) << 1;               // 0 or 2

    const float* xrow = x + (size_t)(m0 + n) * CH + koff;
    const float* wrow = W + (size_t)n * CH + koff;

    v8f c = {};
#pragma unroll 4
    for (int ks = 0; ks < CH / 4; ++ks) {
        v2f a = *(const v2f*)(xrow + 4 * ks);
        v2f b = *(const v2f*)(wrow + 4 * ks);
        c = __builtin_amdgcn_wmma_f32_16x16x4_f32(false, a, false, b,
                                                  (short)0, c, false, false);
    }

    const float bj   = bias[n];
    const int   row0 = m0 + ((lane >> 4) << 3);
#pragma unroll
    for (int v = 0; v < 8; ++v)
        em[(size_t)(row0 + v) * CL + n] = c[v] + bj;
}

// ---------------------------------------------------------------------------
// Kernel 2: gold-path score per batch
// ---------------------------------------------------------------------------
__global__ __launch_bounds__(256)
void crf_gold_score(const float* __restrict__ em, const float* __restrict__ startT,
                    const float* __restrict__ endT, const float* __restrict__ trans,
                    const int* __restrict__ tags, const uint8_t* __restrict__ mask,
                    float* __restrict__ score) {
    const int b   = blockIdx.x;
    const int tid = threadIdx.x;
    __shared__ float sred[256];
    __shared__ int   scnt[256];

    float acc = 0.0f;
    int   cnt = 0;
    for (int t = tid; t < CT; t += 256) {
        const int tg = tags[b * CT + t];
        const int mv = mask[b * CT + t] ? 1 : 0;
        cnt += mv;
        const float e = em[((size_t)b * CT + t) * CL + tg];
        if (t == 0) {
            acc += startT[tg] + e;
        } else {
            const int pg = tags[b * CT + t - 1];
            acc += (trans[pg * CL + tg] + e) * (float)mv;
        }
    }
    sred[tid] = acc;
    scnt[tid] = cnt;
    __syncthreads();
    for (int s = 128; s > 0; s >>= 1) {
        if (tid < s) { sred[tid] += sred[tid + s]; scnt[tid] += scnt[tid + s]; }
        __syncthreads();
    }
    if (tid == 0) {
        const int seq_end = scnt[0] - 1;
        score[b] = sred[0] + endT[tags[b * CT + seq_end]];
    }
}

// ---------------------------------------------------------------------------
// Kernel 3a: per-chunk log-space transfer matrices via WMMA (chunks 1..63).
// Compose step (logsumexp-matmul) done on the matrix pipe:
//   newP[i][j] = r_i + log( sum_k exp(P[i][k]-r_i) * exp(trans[k][j]) ) + em_j
// with r_i = row max. A-frag = exp(P - rowmax) (built from LDS), B-frag =
// exp(trans) (constant, registers), C = 4 chained V_WMMA_F32_16X16X4_F32.
// C-frag layout (lane (j,h) holds rows v+8h, col j) == what the masked select
// needs, what the LDS write-back needs, and the TRANSPOSED global store
// (Pws[b][c][j][i]) the combine kernel reads as contiguous float4s.
// ---------------------------------------------------------------------------
__global__ __launch_bounds__(32)
void crf_chunk_mats(const float* __restrict__ em, const float* __restrict__ trans,
                    const uint8_t* __restrict__ mask, float* __restrict__ Pws) {
    const int b    = blockIdx.x / (CCH - 1);
    const int c    = 1 + (blockIdx.x % (CCH - 1));
    const int lane = threadIdx.x;
    const int j    = lane & 15;             // column owned (C-frag), A-frag row
    const int h    = lane >> 4;

    __shared__ float Pa[256];
    __shared__ float Pb[256];
    __shared__ float rbuf[16];
    float* Pcur = Pa;
    float* Pnxt = Pb;

    // B-fragments of exp(trans): slice s covers K = 4s + 2h + {0,1}; N = j.
    v2f bf[4];
#pragma unroll
    for (int s = 0; s < 4; ++s) {
        const int k0 = 4 * s + 2 * h;
        bf[s].x = __expf(trans[(k0 + 0) * CL + j]);
        bf[s].y = __expf(trans[(k0 + 1) * CL + j]);
    }
    // raw trans column (for init)
    float tc[8];
#pragma unroll
    for (int k = 0; k < 8; ++k) tc[k] = trans[(8 * h + k) * CL + j];

    const float*   emb = em + (size_t)b * CT * CL;
    const uint8_t* mb  = mask + (size_t)b * CT;
    const int      t0  = c * CS;

    // init P = E_{t0}; creg[v] mirrors C-frag: P[v+8h][j]
    float creg[8];
    {
        const float e0 = emb[(size_t)t0 * CL + j];
        const bool  mv = mb[t0] != 0;
#pragma unroll
        for (int k = 0; k < 8; ++k) {
            const int   i   = 8 * h + k;
            const float val = mv ? (tc[k] + e0) : (i == j ? 0.0f : FNEG);
            creg[k]          = val;
            Pcur[i * 16 + j] = val;
        }
    }
    __syncthreads();

    for (int t = t0 + 1; t < t0 + CS; ++t) {
        const float emj = emb[(size_t)t * CL + j];
        const bool  mv  = mb[t] != 0;

        // 1) row max of row j (both halves compute the same value)
        const float4 R0 = *(const float4*)(Pcur + j * 16 + 0);
        const float4 R1 = *(const float4*)(Pcur + j * 16 + 4);
        const float4 R2 = *(const float4*)(Pcur + j * 16 + 8);
        const float4 R3 = *(const float4*)(Pcur + j * 16 + 12);
        const float rj = fmaxf(
            fmaxf(fmaxf(fmaxf(R0.x, R0.y), fmaxf(R0.z, R0.w)),
                  fmaxf(fmaxf(R1.x, R1.y), fmaxf(R1.z, R1.w))),
            fmaxf(fmaxf(fmaxf(R2.x, R2.y), fmaxf(R2.z, R2.w)),
                  fmaxf(fmaxf(R3.x, R3.y), fmaxf(R3.z, R3.w))));
        if (h == 0) rbuf[j] = rj;

        // 2) A-frags exp(P[j][k]-rj) + 4 chained WMMAs (A row m == j)
        v8f cacc = {};
#pragma unroll
        for (int s = 0; s < 4; ++s) {
            const int k0 = 4 * s + 2 * h;
            const v2f p  = *(const v2f*)(Pcur + j * 16 + k0);
            v2f a;
            a.x = __expf(p.x - rj);
            a.y = __expf(p.y - rj);
            cacc = __builtin_amdgcn_wmma_f32_16x16x4_f32(false, a, false, bf[s],
                                                         (short)0, cacc, false, false);
        }
        __syncthreads();   // rbuf visible; all Pcur reads complete

        // 3) newP rows v+8h, col j: r_i + log(C) + em_j; masked -> keep old
        const float4 Ra = *(const float4*)(rbuf + 8 * h);
        const float4 Rb = *(const float4*)(rbuf + 8 * h + 4);
        const float rrow[8] = {Ra.x, Ra.y, Ra.z, Ra.w, Rb.x, Rb.y, Rb.z, Rb.w};
#pragma unroll
        for (int v = 0; v < 8; ++v) {
            const float nval = rrow[v] + __logf(cacc[v]) + emj;
            const float outv = mv ? nval : creg[v];
            creg[v] = outv;
            Pnxt[(v + 8 * h) * 16 + j] = outv;
        }
        __syncthreads();
        float* tmp = Pcur; Pcur = Pnxt; Pnxt = tmp;
    }

    // store transposed: Pws[((b*CCH + c)*16 + j)*16 + i]; creg IS Pt[j][8h..8h+7]
    float* dst = Pws + (((size_t)b * CCH + c) * CL + j) * CL + 8 * h;
#pragma unroll
    for (int v = 0; v < 8; ++v) dst[v] = creg[v];
}

// ---------------------------------------------------------------------------
// Kernel 3b: combine — chunk-0 vector scan (t=1..31), then apply the 63
// chunk matrices sequentially, then denom = logsumexp(alpha + endT).
// One wave32 per batch; critical path = 31 + 63 dependent steps (vs 2047).
// ---------------------------------------------------------------------------
__global__ __launch_bounds__(32)
void crf_combine(const float* __restrict__ em, const float* __restrict__ startT,
                 const float* __restrict__ endT, const float* __restrict__ trans,
                 const uint8_t* __restrict__ mask, const float* __restrict__ Pws,
                 float* __restrict__ denom) {
    const int b    = blockIdx.x;
    const int lane = threadIdx.x;
    const int j    = lane & 15;
    const int h    = lane >> 4;

    __shared__ float alpha[16];

    float tc[8];
#pragma unroll
    for (int k = 0; k < 8; ++k) tc[k] = trans[(8 * h + k) * CL + j];

    const float* emb = em + (size_t)b * CT * CL;

    if (h == 0) alpha[j] = startT[j] + emb[j];
    __syncthreads();

    // chunk 0: vector scan over t = 1..CS-1
    for (int t = 1; t < CS; ++t) {
        const float emj = emb[(size_t)t * CL + j];
        const float4 A0 = *(const float4*)(alpha + 8 * h);
        const float4 A1 = *(const float4*)(alpha + 8 * h + 4);
        const float  aj = alpha[j];
        const float lse = lse8_halfmerge(A0.x + tc[0], A0.y + tc[1],
                                         A0.z + tc[2], A0.w + tc[3],
                                         A1.x + tc[4], A1.y + tc[5],
                                         A1.z + tc[6], A1.w + tc[7]);
        const float na = (mask[b * CT + t] != 0) ? (lse + emj) : aj;
        __syncthreads();
        if (h == 0) alpha[j] = na;
        __syncthreads();
    }

    // apply chunk matrices: alpha <- alpha (x) P_c   (mask baked into P_c)
    for (int c = 1; c < CCH; ++c) {
        const float* pt = Pws + (((size_t)b * CCH + c) * CL + j) * CL + 8 * h;
        const float4 P0 = *(const float4*)pt;
        const float4 P1 = *(const float4*)(pt + 4);
        const float4 A0 = *(const float4*)(alpha + 8 * h);
        const float4 A1 = *(const float4*)(alpha + 8 * h + 4);
        const float nxt = lse8_halfmerge(A0.x + P0.x, A0.y + P0.y,
                                         A0.z + P0.z, A0.w + P0.w,
                                         A1.x + P1.x, A1.y + P1.y,
                                         A1.z + P1.z, A1.w + P1.w);
        __syncthreads();
        if (h == 0) alpha[j] = nxt;
        __syncthreads();
    }

    // denom = logsumexp_j(alpha[j] + endT[j]); widest shuffle first so the
    // inactive upper half merges with live lanes before narrow offsets.
    float m = (h == 0) ? (alpha[j] + endT[j]) : FNEG;
    float s = (h == 0) ? 1.0f : 0.0f;
#pragma unroll
    for (int off = 16; off >= 1; off >>= 1) {
        const float om = __shfl_xor(m, off);
        const float os = __shfl_xor(s, off);
        const float M  = fmaxf(m, om);
        s = s * __expf(m - M) + os * __expf(om - M);
        m = M;
    }
    if (lane == 0) denom[b] = m + __logf(s);
}

// ---------------------------------------------------------------------------
// Kernel 4: out = -mean(score - denom)
// ---------------------------------------------------------------------------
__global__ __launch_bounds__(64)
void crf_final(const float* __restrict__ score, const float* __restrict__ denom,
               float* __restrict__ out) {
    __shared__ float red[64];
    const int t = threadIdx.x;
    red[t] = score[t] - denom[t];
    __syncthreads();
    for (int s = 32; s > 0; s >>= 1) {
        if (t < s) red[t] += red[t + s];
        __syncthreads();
    }
    if (t == 0) out[0] = -red[0] / (float)CB;
}

// ---------------------------------------------------------------------------
// Launch
// ---------------------------------------------------------------------------
extern "C" void kernel_launch(void* const* d_in, const int* in_sizes, int n_in,
                              void* d_out, int out_size, void* d_ws, size_t ws_size,
                              hipStream_t stream) {
    const float*   x      = (const float*)d_in[0];
    const float*   W      = (const float*)d_in[1];
    const float*   bias   = (const float*)d_in[2];
    const float*   startT = (const float*)d_in[3];
    const float*   endT   = (const float*)d_in[4];
    const float*   trans  = (const float*)d_in[5];
    const int*     tags   = (const int*)d_in[6];
    const uint8_t* mask   = (const uint8_t*)d_in[7];
    float*         out    = (float*)d_out;

    float* em    = (float*)d_ws;                        // B*T*L f32   = 8 MiB
    float* Pws   = em + (size_t)CB * CT * CL;           // B*CCH*256   = 4 MiB
    float* score = Pws + (size_t)CB * CCH * CL * CL;    // B f32
    float* denom = score + CB;                          // B f32

    // 1) emissions via fp32 WMMA
    crf_emissions_wmma<<<(CB * CT) / (16 * 8), 256, 0, stream>>>(x, W, bias, em);

    // 2) gold-path score
    crf_gold_score<<<CB, 256, 0, stream>>>(em, startT, endT, trans, tags, mask, score);

    // 3) parallel associative scan: WMMA chunk transfer matrices, then combine
    crf_chunk_mats<<<CB * (CCH - 1), 32, 0, stream>>>(em, trans, mask, Pws);
    crf_combine<<<CB, 32, 0, stream>>>(em, startT, endT, trans, mask, Pws, denom);

    // 4) final scalar
    crf_final<<<1, 64, 0, stream>>>(score, denom, out);
}